// Seq2Seq_LSTM_69303592288393
// MI455X (gfx1250) — compile-verified
//
#include <hip/hip_runtime.h>
#include <hip/hip_bf16.h>

// ---------------------------------------------------------------------------
// Seq2Seq LSTM + attention + vocab head for MI455X (gfx1250, wave32).
//
//  1) gates_x (WMMA f32): G = X @ Wx^T + b  (2048x512x512, both enc & dec)
//  2) lstm_scan: serial recurrence, Wh rows pinned in VGPRs, h in LDS.
//  3) attention: per decoder step fused softmax(encH@h)@encH -> Z[t][128:256]
//  4) head GEMM (WMMA f32 + TDM): Y = Z(2048x256) @ Wy^T + b.
//     A-tile (64x256 f32) DMA'd to LDS once per block via tensor_load_to_lds
//     with row padding (stride 257 dwords) for bank-conflict-free ds reads.
//  5) row softmax over 32000, in place on d_out.
// ---------------------------------------------------------------------------

typedef __attribute__((ext_vector_type(2))) float v2f;
typedef __attribute__((ext_vector_type(8))) float v8f;
typedef __attribute__((ext_vector_type(4))) unsigned int u32x4;
typedef __attribute__((ext_vector_type(8))) int i32x8;
typedef __attribute__((ext_vector_type(4))) int i32x4;

#define S_LEN 2048
#define T_LEN 2048
#define DIM   512
#define H2    128
#define VOCAB 32000

__device__ __forceinline__ float fast_sigmoid(float x) {
    return 1.0f / (1.0f + __expf(-x));
}
__device__ __forceinline__ float fast_tanh(float x) {
    float e = __expf(-2.0f * fabsf(x));
    float t = (1.0f - e) / (1.0f + e);
    return copysignf(t, x);
}

// ---- 1) G[t][g*128+r] = W_g[r,0:512] @ x_t + b_g[r] via WMMA f32 16x16x4 ----
// wave tile: 64(M=t) x 16(N=gate-unit); 32 M-blocks x 32 N-tiles = 1024 waves
__global__ __launch_bounds__(256) void gates_x_wmma_kernel(
    const float* __restrict__ X,                           // (2048, 512)
    const float* __restrict__ Wf, const float* __restrict__ Wi,
    const float* __restrict__ Wo, const float* __restrict__ Wg,  // (128, 640)
    const float* __restrict__ bf, const float* __restrict__ bi,
    const float* __restrict__ bo, const float* __restrict__ bg,
    float* __restrict__ G)                                 // (2048, 512)
{
    const int wave = (blockIdx.x * blockDim.x + threadIdx.x) >> 5;
    const int lane = threadIdx.x & 31;
    const int mt = wave & 31;                              // 32 M-blocks of 64
    const int nt = wave >> 5;                              // 32 N-tiles of 16
    const int m0 = mt * 64, n0 = nt * 16;
    const int lm = lane & 15, hi = lane >> 4;

    const int gi = n0 >> 7;                                // 16-wide tile never crosses a gate
    const int r  = (n0 & 127) + lm;
    const float* W = (gi == 0) ? Wf : (gi == 1) ? Wi : (gi == 2) ? Wo : Wg;
    const float* B = (gi == 0) ? bf : (gi == 1) ? bi : (gi == 2) ? bo : bg;

    const float bias = B[r];
    v8f acc0, acc1, acc2, acc3;
#pragma unroll
    for (int q = 0; q < 8; ++q) { acc0[q] = bias; acc1[q] = bias; acc2[q] = bias; acc3[q] = bias; }

    const float* arow = X + (m0 + lm) * DIM + 2 * hi;      // A: X rows
    const float* brow = W + r * (DIM + H2) + 2 * hi;       // B[k][n] = W[r(n)][k], x-part

    for (int k0 = 0; k0 < DIM; k0 += 4) {
        v2f b;  b.x  = brow[k0];            b.y  = brow[k0 + 1];
        v2f a0; a0.x = arow[k0];            a0.y = arow[k0 + 1];
        v2f a1; a1.x = arow[16 * DIM + k0]; a1.y = arow[16 * DIM + k0 + 1];
        v2f a2; a2.x = arow[32 * DIM + k0]; a2.y = arow[32 * DIM + k0 + 1];
        v2f a3; a3.x = arow[48 * DIM + k0]; a3.y = arow[48 * DIM + k0 + 1];
        acc0 = __builtin_amdgcn_wmma_f32_16x16x4_f32(false, a0, false, b, (short)0, acc0, false, false);
        acc1 = __builtin_amdgcn_wmma_f32_16x16x4_f32(false, a1, false, b, (short)0, acc1, false, false);
        acc2 = __builtin_amdgcn_wmma_f32_16x16x4_f32(false, a2, false, b, (short)0, acc2, false, false);
        acc3 = __builtin_amdgcn_wmma_f32_16x16x4_f32(false, a3, false, b, (short)0, acc3, false, false);
    }
    const int col = n0 + lm;
#pragma unroll
    for (int q = 0; q < 8; ++q) {
        const int rr = q + 8 * hi;
        G[(m0 +  0 + rr) * 512 + col] = acc0[q];
        G[(m0 + 16 + rr) * 512 + col] = acc1[q];
        G[(m0 + 32 + rr) * 512 + col] = acc2[q];
        G[(m0 + 48 + rr) * 512 + col] = acc3[q];
    }
}

// ---- 2) serial LSTM scan: one workgroup, Wh rows held in VGPRs ----
__global__ __launch_bounds__(512, 1) void lstm_scan_kernel(
    const float* __restrict__ G,                           // (T, 512)
    const float* __restrict__ Wf, const float* __restrict__ Wi,
    const float* __restrict__ Wo, const float* __restrict__ Wg,
    float* __restrict__ Hout,                              // (T, 128)
    float* __restrict__ hc_final,                          // 256 floats
    int T, int emit_pre, int load_init)
{
    __shared__ float h_lds[H2];
    __shared__ float gbuf[4][H2];
    const int tid = threadIdx.x;                           // 0..511
    const int gi = tid >> 7, r = tid & 127;
    const float* W = (gi == 0) ? Wf : (gi == 1) ? Wi : (gi == 2) ? Wo : Wg;
    const float* wrow = W + r * (DIM + H2) + DIM;          // Wh part
    float wreg[H2];
#pragma unroll
    for (int k = 0; k < H2; ++k) wreg[k] = wrow[k];

    float c = 0.0f;
    if (tid < H2) {
        float h0 = 0.0f;
        if (load_init) { h0 = hc_final[tid]; c = hc_final[H2 + tid]; }
        h_lds[tid] = h0;
    }
    __syncthreads();

    for (int t = 0; t < T; ++t) {
        if (emit_pre && tid < H2) Hout[t * H2 + tid] = h_lds[tid];
        float acc = G[t * 512 + tid];
#pragma unroll
        for (int k = 0; k < H2; k += 4) {
            float4 hv = *(const float4*)&h_lds[k];
            acc = fmaf(wreg[k + 0], hv.x, acc);
            acc = fmaf(wreg[k + 1], hv.y, acc);
            acc = fmaf(wreg[k + 2], hv.z, acc);
            acc = fmaf(wreg[k + 3], hv.w, acc);
        }
        float act = (gi == 3) ? fast_tanh(acc) : fast_sigmoid(acc);
        gbuf[gi][r] = act;
        __syncthreads();
        if (tid < H2) {
            float f = gbuf[0][r], i = gbuf[1][r], o = gbuf[2][r], g = gbuf[3][r];
            c = fmaf(f, c, i * g);
            float hn = o * fast_tanh(c);
            h_lds[r] = hn;
            if (!emit_pre) Hout[t * H2 + r] = hn;
        }
        __syncthreads();
    }
    if (emit_pre && tid < H2) { hc_final[tid] = h_lds[tid]; hc_final[H2 + tid] = c; }
}

// ---- 3) attention per decoder step: Z[t] = [h_t, softmax(encH@h_t)@encH] ----
__global__ __launch_bounds__(256) void attention_kernel(
    const float* __restrict__ encH,                        // (2048, 128)
    const float* __restrict__ decH,                        // (2048, 128)
    float* __restrict__ Z)                                 // (2048, 256)
{
    __shared__ float ht[H2];
    __shared__ float sc[S_LEN];
    __shared__ float red[256];
    __shared__ float avp[2][H2];
    const int t = blockIdx.x, tid = threadIdx.x;

    if (tid < H2) ht[tid] = decH[t * H2 + tid];
    __syncthreads();

    float lmax = -3.4e38f;
#pragma unroll
    for (int j = 0; j < 8; ++j) {
        const int s = tid + j * 256;
        const float* e = encH + s * H2;
        float acc = 0.0f;
#pragma unroll
        for (int k = 0; k < H2; k += 4) {
            float4 ev = *(const float4*)(e + k);
            float4 hv = *(const float4*)&ht[k];
            acc = fmaf(ev.x, hv.x, acc); acc = fmaf(ev.y, hv.y, acc);
            acc = fmaf(ev.z, hv.z, acc); acc = fmaf(ev.w, hv.w, acc);
        }
        sc[s] = acc;
        lmax = fmaxf(lmax, acc);
    }
    red[tid] = lmax; __syncthreads();
    for (int o = 128; o > 0; o >>= 1) {
        if (tid < o) red[tid] = fmaxf(red[tid], red[tid + o]);
        __syncthreads();
    }
    const float m = red[0];
    __syncthreads();

    float lsum = 0.0f;
#pragma unroll
    for (int j = 0; j < 8; ++j) {
        const int s = tid + j * 256;
        float e = __expf(sc[s] - m);
        sc[s] = e;
        lsum += e;
    }
    red[tid] = lsum; __syncthreads();
    for (int o = 128; o > 0; o >>= 1) {
        if (tid < o) red[tid] += red[tid + o];
        __syncthreads();
    }
    const float inv = 1.0f / red[0];

    const int j = tid & 127, half = tid >> 7;
    const float* eb = encH + half * 1024 * H2 + j;
    const float* sp = &sc[half * 1024];
    float acc = 0.0f;
    for (int s = 0; s < 1024; ++s) acc = fmaf(sp[s], eb[s * H2], acc);
    avp[half][j] = acc;
    __syncthreads();
    if (tid < H2) {
        Z[t * 256 + tid]      = ht[tid];
        Z[t * 256 + H2 + tid] = (avp[0][tid] + avp[1][tid]) * inv;
    }
}

// ---- 4) head GEMM: WMMA f32 16x16x4 + TDM A-tile staging in LDS ----
// block = 8 waves: shared A-tile (64 rows x 256 K, LDS stride 257 dwords),
// wave w handles N subtile n0 + 16*w; 4 M-subtiles of 16 per wave.
#define A_STRIDE 257
__global__ __launch_bounds__(256) void head_gemm_kernel(
    const float* __restrict__ Zb,                          // (2048, 256)
    const float* __restrict__ Wy,                          // (32000, 256)
    const float* __restrict__ by,                          // (32000)
    float* __restrict__ Y)                                 // (2048, 32000)
{
    __shared__ float alds[64 * A_STRIDE];                  // 65792 B
    const int n0 = blockIdx.x * 128;                       // 250 N-blocks
    const int m0 = blockIdx.y * 64;                        // 32  M-blocks
    const int w    = threadIdx.x >> 5;
    const int lane = threadIdx.x & 31;
    const int lm = lane & 15, hi = lane >> 4;

    if (threadIdx.x < 32) {
        // Tensor DMA: 64 rows x 256 dwords, +1 dword LDS pad per row (stride 257)
        unsigned long long ga = (unsigned long long)(uintptr_t)(Zb + (size_t)m0 * 256);
        u32x4 g0;
        g0[0] = 1u;                                        // count=1, user mode
        g0[1] = (unsigned int)(uintptr_t)alds;             // lds_addr (bytes)
        g0[2] = (unsigned int)(ga & 0xffffffffu);          // global_addr[31:0]
        g0[3] = (unsigned int)((ga >> 32) & 0x01ffffffu)   // global_addr[56:32]
              | (2u << 30);                                // type=2 (image)
        i32x8 g1;
        g1[0] = (int)((2u << 16)                           // data_size = 4 bytes
                    | (1u << 20)                           // pad_enable
                    | (7u << 22));                         // pad_interval: 256 dwords
                                                           // pad_amount code 0 = 1 dword
        g1[1] = (int)(256u << 16);                         // tensor_dim0 = 256
        g1[2] = (int)(64u  << 16);                         // tensor_dim1 = 64
        g1[3] = (int)(256u << 16);                         // tile_dim0 = 256
        g1[4] = 64;                                        // tile_dim1 = 64
        g1[5] = 256;                                       // tensor_dim0_stride = 256
        g1[6] = 0; g1[7] = 0;
        i32x4 gz4 = {0, 0, 0, 0};
        i32x8 gz8 = {0, 0, 0, 0, 0, 0, 0, 0};
        __builtin_amdgcn_tensor_load_to_lds(g0, g1, gz4, gz4, gz8, 0);
        __builtin_amdgcn_s_wait_tensorcnt(0);
    }
    __syncthreads();

    const int nw = n0 + w * 16;
    const float bias = by[nw + lm];
    v8f acc0, acc1, acc2, acc3;
#pragma unroll
    for (int q = 0; q < 8; ++q) { acc0[q] = bias; acc1[q] = bias; acc2[q] = bias; acc3[q] = bias; }

    const float* brow = Wy + (nw + lm) * 256 + 2 * hi;     // B[k][n] = Wy[n][k]
    const float* ap   = alds + lm * A_STRIDE + 2 * hi;     // A from padded LDS tile

    for (int k0 = 0; k0 < 256; k0 += 4) {
        v2f b;  b.x  = brow[k0];                     b.y  = brow[k0 + 1];
        v2f a0; a0.x = ap[ 0 * 16 * A_STRIDE + k0];  a0.y = ap[ 0 * 16 * A_STRIDE + k0 + 1];
        v2f a1; a1.x = ap[ 1 * 16 * A_STRIDE + k0];  a1.y = ap[ 1 * 16 * A_STRIDE + k0 + 1];
        v2f a2; a2.x = ap[ 2 * 16 * A_STRIDE + k0];  a2.y = ap[ 2 * 16 * A_STRIDE + k0 + 1];
        v2f a3; a3.x = ap[ 3 * 16 * A_STRIDE + k0];  a3.y = ap[ 3 * 16 * A_STRIDE + k0 + 1];
        acc0 = __builtin_amdgcn_wmma_f32_16x16x4_f32(false, a0, false, b, (short)0, acc0, false, false);
        acc1 = __builtin_amdgcn_wmma_f32_16x16x4_f32(false, a1, false, b, (short)0, acc1, false, false);
        acc2 = __builtin_amdgcn_wmma_f32_16x16x4_f32(false, a2, false, b, (short)0, acc2, false, false);
        acc3 = __builtin_amdgcn_wmma_f32_16x16x4_f32(false, a3, false, b, (short)0, acc3, false, false);
    }

    const int col = nw + lm;
#pragma unroll
    for (int q = 0; q < 8; ++q) {
        const int rr = q + 8 * hi;
        Y[(size_t)(m0 +  0 + rr) * VOCAB + col] = acc0[q];
        Y[(size_t)(m0 + 16 + rr) * VOCAB + col] = acc1[q];
        Y[(size_t)(m0 + 32 + rr) * VOCAB + col] = acc2[q];
        Y[(size_t)(m0 + 48 + rr) * VOCAB + col] = acc3[q];
    }
}

// ---- 5) in-place row softmax over the vocab ----
__global__ __launch_bounds__(256) void softmax_rows_kernel(float* __restrict__ Y)
{
    __shared__ float red[256];
    const int t = blockIdx.x, tid = threadIdx.x;
    float* row = Y + (size_t)t * VOCAB;

    float lmax = -3.4e38f;
    for (int v = tid; v < VOCAB; v += 256) lmax = fmaxf(lmax, row[v]);
    red[tid] = lmax; __syncthreads();
    for (int o = 128; o > 0; o >>= 1) {
        if (tid < o) red[tid] = fmaxf(red[tid], red[tid + o]);
        __syncthreads();
    }
    const float m = red[0];
    __syncthreads();

    float lsum = 0.0f;
    for (int v = tid; v < VOCAB; v += 256) {
        float e = __expf(row[v] - m);
        row[v] = e;
        lsum += e;
    }
    red[tid] = lsum; __syncthreads();
    for (int o = 128; o > 0; o >>= 1) {
        if (tid < o) red[tid] += red[tid + o];
        __syncthreads();
    }
    const float inv = 1.0f / red[0];
    for (int v = tid; v < VOCAB; v += 256) row[v] *= inv;
}

extern "C" void kernel_launch(void* const* d_in, const int* in_sizes, int n_in,
                              void* d_out, int out_size, void* d_ws, size_t ws_size,
                              hipStream_t stream) {
    const float* src  = (const float*)d_in[0];   // (2048, 512)
    const float* trg  = (const float*)d_in[1];   // (2048, 512)
    const float* Wf_w = (const float*)d_in[2];
    const float* Wf_b = (const float*)d_in[3];
    const float* Wi_w = (const float*)d_in[4];
    const float* Wi_b = (const float*)d_in[5];
    const float* Wo_w = (const float*)d_in[6];
    const float* Wo_b = (const float*)d_in[7];
    const float* Wg_w = (const float*)d_in[8];
    const float* Wg_b = (const float*)d_in[9];
    const float* Wy_w = (const float*)d_in[10];  // (32000, 256)
    const float* Wy_b = (const float*)d_in[11];  // (32000,)
    float* out = (float*)d_out;                  // (2048, 32000)

    float* ws     = (float*)d_ws;
    float* Gx_enc = ws;                              // 2048*512
    float* Gx_dec = Gx_enc + S_LEN * 512;            // 2048*512
    float* encH   = Gx_dec + T_LEN * 512;            // 2048*128
    float* decH   = encH + S_LEN * H2;               // 2048*128
    float* Zb     = decH + T_LEN * H2;               // 2048*256
    float* hcf    = Zb + T_LEN * 256;                // 256

    // 32 M-blocks * 32 N-tiles = 1024 waves; 8 waves per block
    gates_x_wmma_kernel<<<128, 256, 0, stream>>>(
        src, Wf_w, Wi_w, Wo_w, Wg_w, Wf_b, Wi_b, Wo_b, Wg_b, Gx_enc);
    gates_x_wmma_kernel<<<128, 256, 0, stream>>>(
        trg, Wf_w, Wi_w, Wo_w, Wg_w, Wf_b, Wi_b, Wo_b, Wg_b, Gx_dec);

    lstm_scan_kernel<<<1, 512, 0, stream>>>(
        Gx_enc, Wf_w, Wi_w, Wo_w, Wg_w, encH, hcf, S_LEN, /*emit_pre=*/1, /*load_init=*/0);
    lstm_scan_kernel<<<1, 512, 0, stream>>>(
        Gx_dec, Wf_w, Wi_w, Wo_w, Wg_w, decH, hcf, T_LEN, /*emit_pre=*/0, /*load_init=*/1);

    attention_kernel<<<T_LEN, 256, 0, stream>>>(encH, decH, Zb);

    head_gemm_kernel<<<dim3(250, 32), 256, 0, stream>>>(Zb, Wy_w, Wy_b, out);

    softmax_rows_kernel<<<T_LEN, 256, 0, stream>>>(out);
}